// GCN_9483287789581
// MI455X (gfx1250) — compile-verified
//
#include <hip/hip_runtime.h>

typedef float v2f __attribute__((ext_vector_type(2)));
typedef float v8f __attribute__((ext_vector_type(8)));

#define WAVE 32

// ---------------------------------------------------------------------------
// Degree / normalization kernels
// ---------------------------------------------------------------------------
__global__ void deg_init_kernel(float* __restrict__ deg, int n) {
    int i = blockIdx.x * blockDim.x + threadIdx.x;
    if (i < n) deg[i] = 1.0f;   // self-loop contributes 1
}

__global__ void deg_count_kernel(const int* __restrict__ dst, float* __restrict__ deg, int nE) {
    int e = blockIdx.x * blockDim.x + threadIdx.x;
    if (e < nE) unsafeAtomicAdd(&deg[dst[e]], 1.0f);
}

__global__ void dinv_kernel(float* __restrict__ deg, int n) {
    int i = blockIdx.x * blockDim.x + threadIdx.x;
    if (i < n) deg[i] = rsqrtf(deg[i]);   // deg >= 1 always (self-loops)
}

// ---------------------------------------------------------------------------
// WMMA fp32 GEMM: H[n_rows x F] = X[n_rows x 128] * W[128 x F]
// Block = 16 output rows; wave w computes the 16x16 tile at cols [16w, 16w+16).
// A tile and transposed W panel staged in LDS with 132-float padded stride so
// each half-wave's b64 pair loads are bank-conflict-free:
//   bank(lane m, dword k) = (132*m + k) mod 64 = (4m + k) mod 64.
// Inner loop: ds_load_b64 (A pair) + ds_load_b64 (B pair) + v_wmma_f32_16x16x4_f32,
// all offsets compile-time immediates (F is a template parameter).
// ---------------------------------------------------------------------------
template <int F>
__global__ void gemm_wmma_f32_kernel(const float* __restrict__ X,
                                     const float* __restrict__ W,
                                     float* __restrict__ H,
                                     int n_rows) {
    constexpr int K   = 128;
    constexpr int LDP = 132;                    // padded LDS row stride (floats)
    __shared__ float tileA[16 * LDP];           // X rows  [row][k]
    __shared__ float tileB[F * LDP];            // W transposed [col][k]

    const int row0 = blockIdx.x * 16;
    const int tid  = threadIdx.x;

    // Stage 16 x 128 A-tile (clamp rows for safety at the tail).
    for (int i = tid; i < 16 * K; i += blockDim.x) {
        int r = i >> 7;                         // i / 128
        int k = i & (K - 1);
        int rr = row0 + r; if (rr >= n_rows) rr = n_rows - 1;
        tileA[r * LDP + k] = X[(size_t)rr * K + k];
    }
    // Stage W transposed: tileB[c][k] = W[k][c]   (coalesced global reads).
    for (int i = tid; i < K * F; i += blockDim.x) {
        int k = i / F;                          // F is 2^n -> shift
        int c = i & (F - 1);
        tileB[c * LDP + k] = W[i];
    }
    __syncthreads();

    const int wave = tid >> 5;
    const int lane = tid & 31;
    const int col0 = wave * 16;
    if (col0 >= F) return;                      // whole wave exits together

    const int m    = lane & 15;                 // A row / B col within tile
    const int koff = (lane >> 4) << 1;          // lanes 0-15: K{0,1}; 16-31: K{2,3}
    const float* Ap = &tileA[m * LDP + koff];
    const float* Bp = &tileB[(col0 + m) * LDP + koff];

    v8f c = {};
    #pragma unroll
    for (int k0 = 0; k0 < K; k0 += 4) {
        v2f a = { Ap[k0], Ap[k0 + 1] };
        v2f b = { Bp[k0], Bp[k0 + 1] };
        c = __builtin_amdgcn_wmma_f32_16x16x4_f32(
                /*neg_a=*/false, a, /*neg_b=*/false, b,
                /*c_mod=*/(short)0, c, /*reuse_a=*/false, /*reuse_b=*/false);
    }

    // C/D layout: VGPR r -> row r (lanes 0-15) / row r+8 (lanes 16-31), col = lane&15.
    const int rbase = (lane >> 4) << 3;
    #pragma unroll
    for (int r = 0; r < 8; ++r) {
        int row = row0 + rbase + r;
        if (row < n_rows) H[(size_t)row * F + col0 + m] = c[r];
    }
}

// ---------------------------------------------------------------------------
// agg[i, c] = bias[c]  (broadcast init; scatter adds on top)
// ---------------------------------------------------------------------------
__global__ void init_bias_kernel(float* __restrict__ agg, const float* __restrict__ bias,
                                 int Fmask, size_t total) {
    size_t i = (size_t)blockIdx.x * blockDim.x + threadIdx.x;
    if (i < total) agg[i] = bias[(int)(i & (size_t)Fmask)];
}

// ---------------------------------------------------------------------------
// Edge scatter: one wave per edge.  Edges [0,nE) are real (src/dst arrays),
// edges [nE, nE+nN) are the self-loops (src = dst = e - nE).
//   agg[dst, :] += h[src, :] * dinv[src] * dinv[dst]
// F = 128 -> float4 per lane (4 atomics); F = 64 -> float2 per lane (2 atomics).
// Working set is ~L2-resident (192 MB), so atomics land in L2.
// ---------------------------------------------------------------------------
template <int F>
__global__ void scatter_kernel(const float* __restrict__ h,
                               const int* __restrict__ src,
                               const int* __restrict__ dst,
                               const float* __restrict__ dinv,
                               float* __restrict__ agg,
                               int nE, int nN) {
    const long long gid  = (long long)blockIdx.x * blockDim.x + threadIdx.x;
    const int       edge = (int)(gid >> 5);
    const int       lane = (int)(gid & 31);
    if (edge >= nE + nN) return;

    int s, d;
    if (edge < nE) { s = src[edge]; d = dst[edge]; }
    else           { s = d = edge - nE; }

    const float w = dinv[s] * dinv[d];
    constexpr int C = F / WAVE;                 // elements per lane (4 or 2)
    const float* hp = h   + (size_t)s * F + lane * C;
    float*       ap = agg + (size_t)d * F + lane * C;

    if constexpr (C == 4) {
        float4 v = *reinterpret_cast<const float4*>(hp);
        unsafeAtomicAdd(ap + 0, v.x * w);
        unsafeAtomicAdd(ap + 1, v.y * w);
        unsafeAtomicAdd(ap + 2, v.z * w);
        unsafeAtomicAdd(ap + 3, v.w * w);
    } else {
        float2 v = *reinterpret_cast<const float2*>(hp);
        unsafeAtomicAdd(ap + 0, v.x * w);
        unsafeAtomicAdd(ap + 1, v.y * w);
    }
}

__global__ void relu_kernel(float* __restrict__ a, size_t total) {
    size_t i = (size_t)blockIdx.x * blockDim.x + threadIdx.x;
    if (i < total) a[i] = fmaxf(a[i], 0.0f);
}

// ---------------------------------------------------------------------------
// Launch
// ---------------------------------------------------------------------------
extern "C" void kernel_launch(void* const* d_in, const int* in_sizes, int n_in,
                              void* d_out, int out_size, void* d_ws, size_t ws_size,
                              hipStream_t stream) {
    const float* x  = (const float*)d_in[0];
    const int*   ei = (const int*)  d_in[1];
    const float* W1 = (const float*)d_in[2];
    const float* b1 = (const float*)d_in[3];
    const float* W2 = (const float*)d_in[4];
    const float* b2 = (const float*)d_in[5];
    float*       out = (float*)d_out;

    const int IN_C  = 128;
    const int HID_C = 128;
    const int OUT_C = in_sizes[5];            // 64
    const int N     = in_sizes[0] / IN_C;     // 100000
    const int E     = in_sizes[1] / 2;        // 1600000

    const int* src = ei;                      // edge_index[0]
    const int* dst = ei + E;                  // edge_index[1]

    // Workspace partition (floats)
    float* ws   = (float*)d_ws;
    size_t o    = 0;
    float* dinv = ws + o;  o += ((size_t)N + 511) & ~(size_t)511;
    float* h1   = ws + o;  o += (size_t)N * HID_C;
    float* agg1 = ws + o;  o += (size_t)N * HID_C;
    float* h2   = ws + o;  // N * OUT_C

    const int TPB = 256;
    const int nEdges = E + N;
    const int scatterBlocks = (int)(((long long)nEdges * WAVE + TPB - 1) / TPB);

    // 1) symmetric normalization: dinv = rsqrt(1 + indeg)
    deg_init_kernel <<<(N + TPB - 1) / TPB, TPB, 0, stream>>>(dinv, N);
    deg_count_kernel<<<(E + TPB - 1) / TPB, TPB, 0, stream>>>(dst, dinv, E);
    dinv_kernel     <<<(N + TPB - 1) / TPB, TPB, 0, stream>>>(dinv, N);

    // 2) layer 1: h1 = x @ W1 (WMMA, 8 waves/block)
    gemm_wmma_f32_kernel<128><<<(N + 15) / 16, 256, 0, stream>>>(x, W1, h1, N);

    // 3) aggregate: agg1 = b1 + scatter(norm * h1), then ReLU
    {
        size_t total = (size_t)N * HID_C;
        init_bias_kernel<<<(int)((total + TPB - 1) / TPB), TPB, 0, stream>>>(agg1, b1, HID_C - 1, total);
        scatter_kernel<128><<<scatterBlocks, TPB, 0, stream>>>(h1, src, dst, dinv, agg1, E, N);
        relu_kernel<<<(int)((total + TPB - 1) / TPB), TPB, 0, stream>>>(agg1, total);
    }

    // 4) layer 2: h2 = relu(agg1) @ W2 (WMMA, F=64 -> 4 waves/block)
    gemm_wmma_f32_kernel<64><<<(N + 15) / 16, 128, 0, stream>>>(agg1, W2, h2, N);

    // 5) out = b2 + scatter(norm * h2)
    {
        size_t total = (size_t)N * OUT_C;
        init_bias_kernel<<<(int)((total + TPB - 1) / TPB), TPB, 0, stream>>>(out, b2, OUT_C - 1, total);
        scatter_kernel<64><<<scatterBlocks, TPB, 0, stream>>>(h2, src, dst, dinv, out, E, N);
    }
}